// TFFastSpeechLengthRegulator_87522843560238
// MI455X (gfx1250) — compile-verified
//
#include <hip/hip_runtime.h>
#include <hip/hip_bf16.h>
#include <stdint.h>

// Problem constants (match the reference).
#define B_    16
#define L_    512
#define H_    384
#define TOUT_ 3584              // L * (DUR_MAX - 1)
#define ROWS_ (B_ * TOUT_)      // 57344

typedef float v4f __attribute__((ext_vector_type(4)));

// ---------------------------------------------------------------------------
// Kernel 1: per-batch inclusive scan of durations (async global->LDS on the
// gfx1250 ASYNCcnt path), then searchsorted(side='right') for every output
// frame. Writes idx (-1 if masked) to workspace (RT: re-read immediately by
// kernel 2 from L2) and the mask floats to the tail of d_out (NT: write-once).
// One block per batch, 512 threads (16 wave32 waves).
// ---------------------------------------------------------------------------
__global__ __launch_bounds__(L_) void lr_scan_idx_kernel(
    const int* __restrict__ dur,      // [B, L]
    int*       __restrict__ idx_ws,   // [B, TOUT]
    float*     __restrict__ mask_out) // [B, TOUT] (tail of d_out, as floats)
{
    __shared__ int cs[L_];

    const int b   = blockIdx.x;
    const int tid = threadIdx.x;

    // --- Async global -> LDS load of this batch's durations (CDNA5 path). ---
    {
        unsigned   lds_addr = (unsigned)(uintptr_t)(&cs[tid]);   // LDS byte addr
        const int* gptr     = dur + b * L_ + tid;                // 64-bit vaddr
        asm volatile("global_load_async_to_lds_b32 %0, %1, off"
                     :
                     : "v"(lds_addr), "v"(gptr)
                     : "memory");
        asm volatile("s_wait_asynccnt 0" ::: "memory");
    }
    __syncthreads();

    // --- Hillis-Steele inclusive scan over 512 elements (9 steps). ---
    #pragma unroll
    for (int off = 1; off < L_; off <<= 1) {
        int v = cs[tid];
        if (tid >= off) v += cs[tid - off];
        __syncthreads();
        cs[tid] = v;
        __syncthreads();
    }

    const int total = cs[L_ - 1];

    // --- Each thread resolves 7 output frames: t = k*512 + tid. ---
    #pragma unroll
    for (int k = 0; k < TOUT_ / L_; ++k) {
        const int t = k * L_ + tid;

        // first j with cs[j] > t  ==  searchsorted(cs, t, side='right')
        int lo = 0, hi = L_;
        #pragma unroll
        for (int s = 0; s < 9; ++s) {           // ceil(log2(512)) = 9
            const int mid = (lo + hi) >> 1;
            if (cs[mid] <= t) lo = mid + 1; else hi = mid;
        }
        int idx = lo;
        if (idx > L_ - 1) idx = L_ - 1;

        const bool live = (t < total);
        idx_ws[b * TOUT_ + t] = live ? idx : -1;          // RT: consumed next
        __builtin_nontemporal_store(live ? 1.0f : 0.0f,   // NT: write-once
                                    &mask_out[b * TOUT_ + t]);
    }
}

// ---------------------------------------------------------------------------
// Kernel 2: bandwidth kernel. One wave per output row; each lane moves three
// float4 (32 lanes x 3 x 16B = 1536B = one H=384 row), so every vmem op is a
// fully coalesced 512B global_load_b128 / global_store_b128 wave transaction.
// Stores are non-temporal (th:TH_STORE_NT): the 88MB output stream is never
// re-read, so it must not displace the reused 12.6MB source tile in L2.
// Masked rows store zeros without touching the source.
// 256 threads/block = 8 wave32 waves = 8 rows per block.
// ---------------------------------------------------------------------------
__global__ __launch_bounds__(256) void lr_gather_kernel(
    const float* __restrict__ src,    // [B, L, H]
    const int*   __restrict__ idx_ws, // [B, TOUT]
    float*       __restrict__ out)    // [B, TOUT, H]
{
    const int row  = blockIdx.x * 8 + (threadIdx.x >> 5); // one wave per row
    const int lane = threadIdx.x & 31;

    const int idx = idx_ws[row];                // lane-uniform
    v4f* dst = reinterpret_cast<v4f*>(out + (size_t)row * H_) + lane;

    if (idx < 0) {
        const v4f z = {0.f, 0.f, 0.f, 0.f};
        __builtin_nontemporal_store(z, dst);
        __builtin_nontemporal_store(z, dst + 32);
        __builtin_nontemporal_store(z, dst + 64);
    } else {
        const int b = row / TOUT_;              // const divisor -> mul/shift
        const v4f* s =
            reinterpret_cast<const v4f*>(src + ((size_t)b * L_ + idx) * H_) + lane;
        v4f a0 = s[0];                          // RT loads: row reused ~3.5x
        v4f a1 = s[32];
        v4f a2 = s[64];
        __builtin_nontemporal_store(a0, dst);
        __builtin_nontemporal_store(a1, dst + 32);
        __builtin_nontemporal_store(a2, dst + 64);
    }
}

// ---------------------------------------------------------------------------
// Host-side launcher.
// d_in[0] = encoder_hidden_states (float32, B*L*H)
// d_in[1] = durations_gt          (int32,   B*L)
// d_in[2] = max_durations         (scalar; compile-time constant here)
// d_out   = [outputs (B*TOUT*H) | mask (B*TOUT)] as float32
// d_ws    = idx scratch (B*TOUT int32 = 224 KiB)
// ---------------------------------------------------------------------------
extern "C" void kernel_launch(void* const* d_in, const int* in_sizes, int n_in,
                              void* d_out, int out_size, void* d_ws, size_t ws_size,
                              hipStream_t stream)
{
    const float* ehs = (const float*)d_in[0];
    const int*   dur = (const int*)d_in[1];

    float* out      = (float*)d_out;
    float* mask_out = out + (size_t)ROWS_ * H_;
    int*   idx_ws   = (int*)d_ws;

    lr_scan_idx_kernel<<<B_, L_, 0, stream>>>(dur, idx_ws, mask_out);
    lr_gather_kernel<<<ROWS_ / 8, 256, 0, stream>>>(ehs, idx_ws, out);
}